// HNM_propmap_32521492365974
// MI455X (gfx1250) — compile-verified
//
#include <hip/hip_runtime.h>
#include <stdint.h>

// ---- problem constants (match reference) ----
#define B_ 8
#define C_ 32
#define W_ 32
#define H_ 32
#define A_ 6
#define NCLS_ 14
#define M_ 64
#define CH_ (3 + NCLS_)                 // 17
#define NCELL (B_*C_*W_*H_*A_)          // 1,572,864
#define NELEM ((long long)NCELL * CH_)  // 26,738,688
#define HARD_NUM_ 256
#define LAM_HNM_ 0.2f
#define LAM_NOOBJ_ 0.001f

// ---- streaming tile: 256 cells x 17 floats = 17408 B (16B aligned) ----
#define CHUNK_CELLS 256
#define CHUNK_FLOATS (CHUNK_CELLS*CH_)  // 4352
#define CHUNK_BYTES  (CHUNK_FLOATS*4)   // 17408
#define CHUNK_B128   (CHUNK_BYTES/16)   // 1088
#define NCHUNK       (NCELL/CHUNK_CELLS)// 6144 exactly (no tail)
#define NSTREAM_BLOCKS 1536             // 4 chunks per block

// ---- histogram select ----
#define NB 512
#define HLO  (-8.0f)
#define HSCL ((float)NB / 16.0f)

// ---- workspace layout (bytes) ----
#define WS_ACC   0      // float[8]: 0 clpos 1 clneg 2 reg 3 P 4 regu 5 hn
#define WS_CNT   64     // int[14] hn counts
#define WS_TOTK  128    // int[1]
#define WS_TB    192    // int[16] threshold bin per class
#define WS_NEED  256    // int[16] remaining take count in boundary bin
#define WS_TAKEN 320    // int[16] atomic take counters
#define WS_HIST  512    // int[14*512]
#define WS_BYTES (512 + NCLS_*NB*4)

typedef __attribute__((ext_vector_type(2))) float v2f;
typedef __attribute__((ext_vector_type(8))) float v8f;

#if __has_builtin(__builtin_amdgcn_s_wait_asynccnt)
#define WAIT_ASYNC0() __builtin_amdgcn_s_wait_asynccnt(0)
#else
#define WAIT_ASYNC0() asm volatile("s_wait_asynccnt 0" ::: "memory")
#endif

__device__ __forceinline__ void async_ld_b128(uint32_t lds_off, uint64_t gaddr) {
  // GV mode: VDST = LDS byte offset, VADDR = 64-bit global address, SADDR = off
  asm volatile("global_load_async_to_lds_b128 %0, %1, off"
               :: "v"(lds_off), "v"(gaddr) : "memory");
}

__device__ __forceinline__ float softplusf(float x) {
  return fmaxf(x, 0.0f) + log1pf(expf(-fabsf(x)));
}

__device__ __forceinline__ int binOf(float v) {
  int b = (int)((v - HLO) * HSCL);
  return b < 0 ? 0 : (b >= NB ? NB - 1 : b);
}

// Exact fp32 wave32 reduction. WMMA trick: D = ones(16x4) x B(4x16) + C gives
// column sums; with B = {p_lane, 0} every lane's D[.,j] = p_j + p_{j+16},
// so one v_wmma replaces one shuffle level; 4 xor-shuffles finish 16->1.
__device__ __forceinline__ float waveReduceSum(float p) {
#if __has_builtin(__builtin_amdgcn_wmma_f32_16x16x4_f32)
  v2f a; a[0] = 1.0f; a[1] = 1.0f;
  v2f b; b[0] = p;    b[1] = 0.0f;
  v8f c = {};
  c = __builtin_amdgcn_wmma_f32_16x16x4_f32(false, a, false, b, (short)0, c, false, false);
  float s = c[0];
  s += __shfl_xor(s, 1, 32);
  s += __shfl_xor(s, 2, 32);
  s += __shfl_xor(s, 4, 32);
  s += __shfl_xor(s, 8, 32);
  return s;
#else
  for (int m = 16; m >= 1; m >>= 1) p += __shfl_xor(p, m, 32);
  return p;
#endif
}

__device__ __forceinline__ void blockAtomicAdd(float v, float* dst, float* wred) {
  float w = waveReduceSum(v);
  const int lane = threadIdx.x & 31, wid = threadIdx.x >> 5;
  if (lane == 0) wred[wid] = w;
  __syncthreads();
  if (threadIdx.x == 0) {
    float s = 0.f;
    for (int i = 0; i < 8; ++i) s += wred[i];
    atomicAdd(dst, s);
  }
}

// ---- K1: gather proposals -> pos/neg/reg sums, P, hn counts ----
__global__ void k_gather(const float* __restrict__ pm, const int* __restrict__ pidx,
                         const float* __restrict__ preg, float* acc, int* cnt) {
  int t = blockIdx.x * blockDim.x + threadIdx.x;
  if (t >= B_*A_*M_) return;
  int b = t / (A_*M_);
  int rem = t % (A_*M_);
  int a = rem / M_;
  const int* pi = pidx + (size_t)t*4;
  int c = pi[0], w = pi[1], h = pi[2], lab = pi[3];
  if (lab == -100) return;
  if (lab < 0) { atomicAdd(&cnt[-1 - lab], 1); return; }
  size_t cell = ((((size_t)b*C_ + c)*W_ + w)*H_ + h)*A_ + a;
  const float* vals = pm + cell*CH_;
  const float* rg = preg + (size_t)t*3;
  float clp = softplusf(-vals[3 + lab]);
  float cln = 0.f;
  #pragma unroll
  for (int k = 0; k < NCLS_; ++k)
    if (k != lab) cln += softplusf(vals[3 + k]);
  float rsum = 0.f;
  #pragma unroll
  for (int i = 0; i < 3; ++i) {
    float d = tanhf(vals[i]) - rg[i];
    float ad = fabsf(d);
    rsum += (ad < 1.f) ? 0.5f*d*d : ad - 0.5f;
  }
  atomicAdd(&acc[0], clp);
  atomicAdd(&acc[1], cln);
  atomicAdd(&acc[2], rsum);
  atomicAdd(&acc[3], 1.0f);
}

// ---- K2: one HBM pass: softplus total + per-class LDS histograms ----
__global__ __launch_bounds__(256) void k_stream_hist(const float* __restrict__ pm,
                                                     float* acc, int* ghist) {
  __shared__ float tileF[CHUNK_FLOATS];
  __shared__ int   hist[NCLS_*NB];
  __shared__ float wred[8];
  const int tid = threadIdx.x;
  for (int i = tid; i < NCLS_*NB; i += 256) hist[i] = 0;
  __syncthreads();
  const uint32_t ldsBase = (uint32_t)(size_t)&tileF[0];
  float regu = 0.f;
  for (int chunk = blockIdx.x; chunk < NCHUNK; chunk += NSTREAM_BLOCKS) {
    const uint64_t gbase = (uint64_t)(size_t)pm + (uint64_t)chunk * CHUNK_BYTES;
    for (int i = tid; i < CHUNK_B128; i += 256)
      async_ld_b128(ldsBase + (uint32_t)i*16u, gbase + (uint64_t)i*16u);
    WAIT_ASYNC0();
    __syncthreads();
    const float* cell = &tileF[tid * CH_];
    float s = 0.f;
    #pragma unroll
    for (int k = 0; k < CH_; ++k) s += softplusf(cell[k]);
    regu += s;
    #pragma unroll
    for (int c = 0; c < NCLS_; ++c)
      atomicAdd(&hist[c*NB + binOf(cell[3 + c])], 1);
    __syncthreads();  // all reads done before next chunk overwrites tile
  }
  for (int i = tid; i < NCLS_*NB; i += 256) {
    int v = hist[i];
    if (v) atomicAdd(&ghist[i], v);
  }
  blockAtomicAdd(regu, &acc[4], wred);
}

// ---- K3: per-class threshold bin from histogram suffix counts ----
__global__ void k_thresh(const int* cnt, const int* ghist,
                         int* tb, int* need, int* taken, int* totk) {
  int c = threadIdx.x;
  if (c == 0) {
    int t = 0;
    for (int i = 0; i < NCLS_; ++i) t += cnt[i] * HARD_NUM_;
    *totk = t;
  }
  if (c < NCLS_) {
    long long k = (long long)cnt[c] * HARD_NUM_;
    int bsel = NB; long long r = 0, cum = 0;
    if (k > 0) {
      for (int b = NB - 1; b >= 0; --b) {
        long long n = ghist[c*NB + b];
        if (cum + n >= k) { bsel = b; r = k - cum; break; }
        cum += n;
      }
      if (bsel == NB) { bsel = -1; r = 0; }  // k >= total: take everything
    }
    tb[c] = bsel;
    need[c] = (int)r;
    taken[c] = 0;
  }
}

// ---- K4: second streaming pass (L2-resident): sum softplus of top-k ----
__global__ __launch_bounds__(256) void k_hn(const float* __restrict__ pm, float* acc,
                                            const int* tb, const int* need, int* taken) {
  __shared__ float tileF[CHUNK_FLOATS];
  __shared__ float wred[8];
  __shared__ int s_tb[NCLS_], s_need[NCLS_];
  const int tid = threadIdx.x;
  if (tid < NCLS_) { s_tb[tid] = tb[tid]; s_need[tid] = need[tid]; }
  __syncthreads();
  const uint32_t ldsBase = (uint32_t)(size_t)&tileF[0];
  float hn = 0.f;
  for (int chunk = blockIdx.x; chunk < NCHUNK; chunk += NSTREAM_BLOCKS) {
    const uint64_t gbase = (uint64_t)(size_t)pm + (uint64_t)chunk * CHUNK_BYTES;
    for (int i = tid; i < CHUNK_B128; i += 256)
      async_ld_b128(ldsBase + (uint32_t)i*16u, gbase + (uint64_t)i*16u);
    WAIT_ASYNC0();
    __syncthreads();
    const float* cell = &tileF[tid * CH_];
    #pragma unroll
    for (int c = 0; c < NCLS_; ++c) {
      float v = cell[3 + c];
      int b = binOf(v);
      int T = s_tb[c];
      if (b > T) {
        hn += softplusf(v);
      } else if (b == T && s_need[c] > 0) {
        int old = atomicAdd(&taken[c], 1);
        if (old < s_need[c]) hn += softplusf(v);
      }
    }
    __syncthreads();
  }
  blockAtomicAdd(hn, &acc[5], wred);
}

// ---- K5: combine terms ----
__global__ void k_final(const float* acc, const int* totk, float* out) {
  if (threadIdx.x == 0 && blockIdx.x == 0) {
    float P = fmaxf(acc[3], 1.0f);
    float loss = acc[0] / P
               + acc[1] / (P * (float)((NCLS_-1)*(NCLS_-1)))
               + acc[2] / (3.0f * P)
               + LAM_NOOBJ_ * (acc[4] / (float)NELEM);
    int tk = *totk;
    if (tk > 0) loss += LAM_HNM_ * acc[5] / (float)(tk > 1 ? tk : 1);
    out[0] = loss;
  }
}

extern "C" void kernel_launch(void* const* d_in, const int* in_sizes, int n_in,
                              void* d_out, int out_size, void* d_ws, size_t ws_size,
                              hipStream_t stream) {
  (void)in_sizes; (void)n_in; (void)out_size; (void)ws_size;
  const float* pm  = (const float*)d_in[0];
  const int*  pidx = (const int*)d_in[1];
  const float* prg = (const float*)d_in[2];
  float* out = (float*)d_out;
  char* ws = (char*)d_ws;
  float* acc  = (float*)(ws + WS_ACC);
  int* cnt    = (int*)(ws + WS_CNT);
  int* totk   = (int*)(ws + WS_TOTK);
  int* tb     = (int*)(ws + WS_TB);
  int* need   = (int*)(ws + WS_NEED);
  int* taken  = (int*)(ws + WS_TAKEN);
  int* ghist  = (int*)(ws + WS_HIST);

  hipMemsetAsync(ws, 0, WS_BYTES, stream);
  k_gather<<<(B_*A_*M_ + 255)/256, 256, 0, stream>>>(pm, pidx, prg, acc, cnt);
  k_stream_hist<<<NSTREAM_BLOCKS, 256, 0, stream>>>(pm, acc, ghist);
  k_thresh<<<1, 64, 0, stream>>>(cnt, ghist, tb, need, taken, totk);
  k_hn<<<NSTREAM_BLOCKS, 256, 0, stream>>>(pm, acc, tb, need, taken);
  k_final<<<1, 1, 0, stream>>>(acc, totk, out);
}